// SAGENet_37082747633734
// MI455X (gfx1250) — compile-verified
//
#include <hip/hip_runtime.h>
#include <hip/hip_bf16.h>

typedef __attribute__((ext_vector_type(2))) float v2f;
typedef __attribute__((ext_vector_type(8))) float v8f;

#define N_NODES 100000
#define N_EDGES 1600000

// ---------------- zero fill ----------------
__global__ __launch_bounds__(256) void k_zero(float* __restrict__ p, int n) {
  int i = blockIdx.x * blockDim.x + threadIdx.x;
  int stride = gridDim.x * blockDim.x;
  for (; i < n; i += stride) p[i] = 0.0f;
}

// ---------------- in-degree (once, reused for all 3 layers) ----------------
__global__ __launch_bounds__(256) void k_degree(const int* __restrict__ dst,
                                                float* __restrict__ deg, int nEdges) {
  int i = blockIdx.x * blockDim.x + threadIdx.x;
  int stride = gridDim.x * blockDim.x;
  for (; i < nEdges; i += stride) atomicAdd(&deg[dst[i]], 1.0f);
}

// ---------------- dense GEMM: Out[N,dout] = H[N,K] @ W[K,dout] ----------------
// One wave computes one 16x16 output tile via V_WMMA_F32_16X16X4_F32.
// A layout (16x4 f32): lanes 0-15 hold K=k0,k0+1 ; lanes 16-31 hold K=k0+2,k0+3
// C/D layout: VGPR r -> M = r + 8*(lane>=16), N = lane&15
__global__ __launch_bounds__(256) void k_gemm_wmma(
    const float* __restrict__ H, const float* __restrict__ W,
    float* __restrict__ Out, int nRows, int K, int dout)
{
  const int lane   = threadIdx.x & 31;
  const int wave   = threadIdx.x >> 5;
  const int rowTile = blockIdx.x * 8 + wave;
  if (rowTile * 16 >= nRows) return;          // wave-uniform: EXEC stays all-1
  const int colTile = blockIdx.y;

  const int m     = lane & 15;
  const int khalf = (lane >> 4) << 1;         // 0 or 2
  const int col   = colTile * 16 + m;

  const float* __restrict__ hrow = H + (size_t)(rowTile * 16 + m) * K + khalf;
  const float* __restrict__ wcol = W + (size_t)khalf * dout + col;

  v8f acc = {};
  for (int k0 = 0; k0 < K; k0 += 4) {
    v2f a = *(const v2f*)(hrow + k0);         // 8B-aligned: K%4==0, khalf even
    v2f b;
    b.x = wcol[(size_t)k0 * dout];
    b.y = wcol[(size_t)(k0 + 1) * dout];
    acc = __builtin_amdgcn_wmma_f32_16x16x4_f32(
        /*neg_a=*/false, a, /*neg_b=*/false, b,
        /*c_mod=*/(short)0, acc, /*reuse_a=*/false, /*reuse_b=*/false);
  }

  const int mbase = (lane >> 4) << 3;         // 0 or 8
  float* __restrict__ orow =
      Out + (size_t)(rowTile * 16 + mbase) * dout + colTile * 16 + m;
#pragma unroll
  for (int r = 0; r < 8; ++r) orow[(size_t)r * dout] = acc[r];
}

// ---------------- edge scatter-add: agg[dst] += hn[src] ----------------
// lane = feature index; a wave32 covers one edge's 32 features (dout=32)
// -> coalesced gather + coalesced global_atomic_add_f32, L2-resident rows.
__global__ __launch_bounds__(256) void k_scatter(
    const float* __restrict__ hn, const int* __restrict__ src,
    const int* __restrict__ dst, float* __restrict__ agg,
    int nEdges, int doutShift)
{
  const int dmask = (1 << doutShift) - 1;
  const int total = nEdges << doutShift;
  int i = blockIdx.x * blockDim.x + threadIdx.x;
  int stride = gridDim.x * blockDim.x;
  for (; i < total; i += stride) {
    int e = i >> doutShift;
    int f = i & dmask;
    float v = hn[((size_t)src[e] << doutShift) + f];
    atomicAdd(&agg[((size_t)dst[e] << doutShift) + f], v);
  }
}

// ---------------- combine: out = [relu](hs + agg/max(deg,1) + b) ----------------
__global__ __launch_bounds__(256) void k_combine(
    const float* __restrict__ hs, const float* __restrict__ agg,
    const float* __restrict__ deg, const float* __restrict__ bias,
    float* __restrict__ out, int nNodes, int doutShift, int doRelu)
{
  const int dmask = (1 << doutShift) - 1;
  const int total = nNodes << doutShift;
  int i = blockIdx.x * blockDim.x + threadIdx.x;
  int stride = gridDim.x * blockDim.x;
  for (; i < total; i += stride) {
    int node = i >> doutShift;
    float d = deg[node];
    float v = hs[i] + agg[i] / fmaxf(d, 1.0f) + bias[i & dmask];
    if (doRelu) v = fmaxf(v, 0.0f);
    out[i] = v;
  }
}

static inline int cdiv(int a, int b) { return (a + b - 1) / b; }

extern "C" void kernel_launch(void* const* d_in, const int* in_sizes, int n_in,
                              void* d_out, int out_size, void* d_ws, size_t ws_size,
                              hipStream_t stream) {
  const float* x    = (const float*)d_in[0];
  const int*   esrc = (const int*)d_in[1];
  const int*   edst = (const int*)d_in[2];
  const float* Ws1  = (const float*)d_in[3];
  const float* Wn1  = (const float*)d_in[4];
  const float* b1   = (const float*)d_in[5];
  const float* Ws2  = (const float*)d_in[6];
  const float* Wn2  = (const float*)d_in[7];
  const float* b2   = (const float*)d_in[8];
  const float* Ws3  = (const float*)d_in[9];
  const float* Wn3  = (const float*)d_in[10];
  const float* b3   = (const float*)d_in[11];

  float* ws  = (float*)d_ws;
  float* deg = ws;                               // N
  float* hs  = deg + N_NODES;                    // N*32
  float* hn  = hs  + (size_t)N_NODES * 32;       // N*32
  float* agg = hn  + (size_t)N_NODES * 32;       // N*32
  float* hA  = agg + (size_t)N_NODES * 32;       // N*32 (layer1 out)
  float* hB  = hA  + (size_t)N_NODES * 32;       // N*32 (layer2 out)

  // degree (computed once, shared by all layers)
  k_zero<<<cdiv(N_NODES, 256), 256, 0, stream>>>(deg, N_NODES);
  k_degree<<<cdiv(N_EDGES, 256), 256, 0, stream>>>(edst, deg, N_EDGES);

  const int nRowTiles = N_NODES / 16;            // 6250, exact

  // -------- layer as a helper --------
  auto layer = [&](const float* hin, int K, int dout,
                   const float* Wself, const float* Wneigh, const float* bias,
                   float* hout, int relu) {
    dim3 gg(cdiv(nRowTiles, 8), dout / 16);
    k_gemm_wmma<<<gg, 256, 0, stream>>>(hin, Wself,  hs, N_NODES, K, dout);
    k_gemm_wmma<<<gg, 256, 0, stream>>>(hin, Wneigh, hn, N_NODES, K, dout);
    int nfeat = N_NODES * dout;
    k_zero<<<cdiv(nfeat, 256), 256, 0, stream>>>(agg, nfeat);
    int shift = (dout == 32) ? 5 : 4;
    int total = N_EDGES * dout;
    k_scatter<<<cdiv(total, 256), 256, 0, stream>>>(hn, esrc, edst, agg, N_EDGES, shift);
    k_combine<<<cdiv(nfeat, 256), 256, 0, stream>>>(hs, agg, deg, bias, hout,
                                                    N_NODES, shift, relu);
  };

  layer(x,  128, 32, Ws1, Wn1, b1, hA, 1);
  layer(hA,  32, 32, Ws2, Wn2, b2, hB, 1);
  layer(hB,  32, 16, Ws3, Wn3, b3, (float*)d_out, 0);
}